// Coefficients_32152125178114
// MI455X (gfx1250) — compile-verified
//
#include <hip/hip_runtime.h>
#include <stdint.h>

typedef float v4f __attribute__((ext_vector_type(4)));

constexpr int E = 4096;          // elements
constexpr int N = 2048;          // nodes
constexpr int C = 2 * E + N;     // 10240 columns; rows = N + 2E = 10240
constexpr int ZROWS = 4;         // rows filled per block in zfill_async

// ---------------------------------------------------------------------------
// Bulk zero-fill of a rectangular region of the output using the CDNA5
// async store-from-LDS path. One 16B zero block staged in LDS is broadcast-
// read by every lane; each async-store instruction streams 512B/wave to HBM
// with no VGPR data operands (tracked on ASYNCcnt). Non-temporal hint keeps
// the 419MB output stream out of L2. Each block covers ZROWS rows to
// amortize the wave-launch / index-math preamble.
// base points at (row0, col0); region is width4*16 bytes wide,
// gridDim.y*ZROWS rows, row pitch = C floats.
// ---------------------------------------------------------------------------
__global__ void zfill_async(float* __restrict__ base, int width4) {
  __shared__ v4f zblk;
  if (threadIdx.x == 0) zblk = (v4f){0.f, 0.f, 0.f, 0.f};
  __syncthreads();
  // LDS aperture: low 32 bits of the flat address are the LDS byte offset.
  unsigned ldsoff = (unsigned)(uintptr_t)&zblk;
  int c4 = blockIdx.x * blockDim.x + threadIdx.x;
  if (c4 < width4) {
    float* p = base + (size_t)blockIdx.y * (ZROWS * (size_t)C) + 4u * (unsigned)c4;
#pragma unroll
    for (int i = 0; i < ZROWS; ++i) {
      asm volatile("global_store_async_from_lds_b128 %0, %1, off th:TH_STORE_NT"
                   :: "v"(p), "v"(ldsoff)
                   : "memory");
      p += C;
    }
  }
#if __has_builtin(__builtin_amdgcn_s_wait_asynccnt)
  __builtin_amdgcn_s_wait_asynccnt(0);
#else
  asm volatile("s_wait_asynccnt 0" ::: "memory");
#endif
}

// ---------------------------------------------------------------------------
// kcl block: out[r, 0:E] = M[r, :], float4 loads + non-temporal float4 stores.
// grid = (E/4/256, N), block = 256
// ---------------------------------------------------------------------------
__global__ void kcl_copy(const float* __restrict__ M, float* __restrict__ out) {
  int c4 = blockIdx.x * blockDim.x + threadIdx.x;   // 0 .. E/4-1
  int r  = blockIdx.y;
  const v4f* src = (const v4f*)(M + (size_t)r * E);
  v4f v = src[c4];
  v4f* dst = (v4f*)(out + (size_t)r * C);
  __builtin_nontemporal_store(v, dst + c4);
}

// ---------------------------------------------------------------------------
// kvl block: out[N+r, 2E+c] = -M[c, r]  via LDS 32x33 tiled transpose.
// Coalesced 128B reads of M rows, coalesced 128B NT writes of output rows.
// grid = (E/32, N/32), block = (32, 8)
// ---------------------------------------------------------------------------
__global__ void kvl_mt(const float* __restrict__ M, float* __restrict__ out) {
  __shared__ float t[32][33];
  int tx = threadIdx.x;
  int r0 = blockIdx.x * 32;   // r index into [0, E)
  int c0 = blockIdx.y * 32;   // c index into [0, N)
  for (int i = 0; i < 32; i += 8) {
    int ty = threadIdx.y + i;
    t[ty][tx] = M[(size_t)(c0 + ty) * E + (r0 + tx)];
  }
  __syncthreads();
  for (int i = 0; i < 32; i += 8) {
    int ty = threadIdx.y + i;
    __builtin_nontemporal_store(
        -t[tx][ty],
        &out[(size_t)(N + r0 + ty) * C + 2 * E + c0 + tx]);
  }
}

// ---------------------------------------------------------------------------
// Diagonals: identity 1.0 in the kvl block, z/y per-element values in the
// element block. sigmoid(p) > 0.5  <=>  p > 0.
// grid = E/256, block = 256
// ---------------------------------------------------------------------------
__global__ void diag_fill(const float* __restrict__ a,
                          const float* __restrict__ params,
                          const float* __restrict__ dtp,
                          const int*   __restrict__ kinds,
                          const int*   __restrict__ modep,
                          float* __restrict__ out) {
  int e = blockIdx.x * blockDim.x + threadIdx.x;
  if (e >= E) return;

  // I_E inside kvl rows
  out[(size_t)(N + e) * C + E + e] = 1.0f;

  float av  = a[e];
  float pv  = params[e];
  float dtv = dtp[0];
  int   k   = kinds[e];
  bool  tr  = (modep[0] == 1);
  float dta = dtv / av;

  float z = 0.f, y = 0.f;
  if (k == 0)              { z = av;                 y = -1.f; }               // R
  else if (k == 1)         { z = 1.f;                y = tr ? -dta : 0.f; }    // L
  else if (k == 2)         { z = tr ? -dta : 0.f;    y = 1.f; }                // C
  else if (k >= 3 && k <= 5){ z = 0.f;               y = 1.f; }                // VS/VG/CC
  else if (k >= 6 && k <= 8){ z = 1.f;               y = 0.f; }                // CS/CG/VC
  else if (k == 9)         { bool cl = pv > 0.f;     z = cl ? 0.f : 1.f;
                                                      y = cl ? 1.f : 0.f; }    // SW

  size_t rb = (size_t)(N + E + e) * C;
  out[rb + e]     = z;
  out[rb + E + e] = y;
}

// ---------------------------------------------------------------------------
// Inputs (setup_inputs order): 0:M f32[N*E], 1:a f32[E], 2:params f32[E],
// 3:dt f32[1], 4:kinds i32[E], 5:mode i32[1].
// Output: f32[(N+2E)*(2E+N)] = f32[10240*10240].
// ---------------------------------------------------------------------------
extern "C" void kernel_launch(void* const* d_in, const int* in_sizes, int n_in,
                              void* d_out, int out_size, void* d_ws, size_t ws_size,
                              hipStream_t stream) {
  (void)in_sizes; (void)n_in; (void)out_size; (void)d_ws; (void)ws_size;
  const float* M      = (const float*)d_in[0];
  const float* a      = (const float*)d_in[1];
  const float* params = (const float*)d_in[2];
  const float* dtp    = (const float*)d_in[3];
  const int*   kinds  = (const int*)d_in[4];
  const int*   modep  = (const int*)d_in[5];
  float* out = (float*)d_out;

  const int T = 256;

  // Zero regions (async store-from-LDS), each element written exactly once.
  // Heights are all multiples of ZROWS(=4).
  // 1) kcl rows, cols [E, C):            2048 rows x 6144 floats
  int w1 = (E + N) / 4;                                    // 1536 float4
  zfill_async<<<dim3((w1 + T - 1) / T, N / ZROWS), T, 0, stream>>>(out + E, w1);
  // 2) kvl rows, cols [0, 2E):           4096 rows x 8192 floats
  int w2 = (2 * E) / 4;                                    // 2048 float4
  zfill_async<<<dim3((w2 + T - 1) / T, E / ZROWS), T, 0, stream>>>(out + (size_t)N * C, w2);
  // 3) element rows, cols [0, C):        4096 rows x 10240 floats
  int w3 = C / 4;                                          // 2560 float4
  zfill_async<<<dim3((w3 + T - 1) / T, E / ZROWS), T, 0, stream>>>(out + (size_t)(N + E) * C, w3);

  // Dense data blocks
  kcl_copy<<<dim3((E / 4 + T - 1) / T, N), T, 0, stream>>>(M, out);
  kvl_mt<<<dim3(E / 32, N / 32), dim3(32, 8), 0, stream>>>(M, out);

  // Diagonals last (stream order guarantees they land after the zero fills)
  diag_fill<<<dim3((E + T - 1) / T), T, 0, stream>>>(a, params, dtp, kinds, modep, out);
}